// KMeansCrossAttentionLayer_20985210208600
// MI455X (gfx1250) — compile-verified
//
#include <hip/hip_runtime.h>
#include <hip/hip_bf16.h>

typedef float v2f __attribute__((ext_vector_type(2)));
typedef float v8f __attribute__((ext_vector_type(8)));
typedef int   v4i __attribute__((ext_vector_type(4)));

#define C 256
#define QN 128
#define EPS 1e-5f
#define MB 32         // points per block (two 16-row WMMA tiles)
#define ASTRIDE 260   // padded LDS row stride (floats): 260 % 64 == 4 -> conflict-free frag loads

// CDNA5 async global->LDS staging (ASYNCcnt path), with safe fallback.
#if defined(__has_builtin)
#if __has_builtin(__builtin_amdgcn_global_load_async_to_lds_b128) && \
    __has_builtin(__builtin_amdgcn_s_wait_asynccnt)
#define USE_ASYNC_LDS 1
#endif
#endif

#ifdef USE_ASYNC_LDS
typedef __attribute__((address_space(1))) v4i* gptr_v4i;   // global int4*
typedef __attribute__((address_space(3))) v4i* lptr_v4i;   // LDS int4*
#endif

// ---------------------------------------------------------------------------
// Block-wide sum reduction over 256 threads.
// ---------------------------------------------------------------------------
__device__ __forceinline__ float block_sum(float v, float* red) {
    int t = threadIdx.x;
    red[t] = v;
    __syncthreads();
    #pragma unroll
    for (int s = 128; s > 0; s >>= 1) {
        if (t < s) red[t] += red[t + s];
        __syncthreads();
    }
    float r = red[0];
    __syncthreads();
    return r;
}

// ---------------------------------------------------------------------------
// Kernel 1: MLP head. One block (256 threads) per cluster row q.
// LN -> Linear+ReLU -> Linear+ReLU -> Linear  ==> emb[q, :]
// ---------------------------------------------------------------------------
__global__ __launch_bounds__(256) void mlp_head_kernel(
    const float* __restrict__ centers,
    const float* __restrict__ ln0w, const float* __restrict__ ln0b,
    const float* __restrict__ W1, const float* __restrict__ b1,
    const float* __restrict__ W2, const float* __restrict__ b2,
    const float* __restrict__ W3, const float* __restrict__ b3,
    float* __restrict__ emb)
{
    __shared__ float xs[C];
    __shared__ float ys[C];
    __shared__ float red[C];
    const int t = threadIdx.x;
    const int q = blockIdx.x;

    float x = centers[q * C + t];
    float mean = block_sum(x, red) * (1.0f / C);
    float d = x - mean;
    float var = block_sum(d * d, red) * (1.0f / C);
    xs[t] = d * rsqrtf(var + EPS) * ln0w[t] + ln0b[t];
    __syncthreads();

    // layer 1: ys = relu(xs @ W1^T + b1)
    {
        float s = b1[t];
        const float4* w = (const float4*)(W1 + (size_t)t * C);
        #pragma unroll 4
        for (int c4 = 0; c4 < C / 4; ++c4) {
            float4 wv = w[c4];
            s += xs[c4 * 4 + 0] * wv.x + xs[c4 * 4 + 1] * wv.y +
                 xs[c4 * 4 + 2] * wv.z + xs[c4 * 4 + 3] * wv.w;
        }
        ys[t] = fmaxf(s, 0.0f);
    }
    __syncthreads();

    // layer 2: xs = relu(ys @ W2^T + b2)
    {
        float s = b2[t];
        const float4* w = (const float4*)(W2 + (size_t)t * C);
        #pragma unroll 4
        for (int c4 = 0; c4 < C / 4; ++c4) {
            float4 wv = w[c4];
            s += ys[c4 * 4 + 0] * wv.x + ys[c4 * 4 + 1] * wv.y +
                 ys[c4 * 4 + 2] * wv.z + ys[c4 * 4 + 3] * wv.w;
        }
        xs[t] = fmaxf(s, 0.0f);
    }
    __syncthreads();

    // layer 3: emb = xs @ W3^T + b3
    {
        float s = b3[t];
        const float4* w = (const float4*)(W3 + (size_t)t * C);
        #pragma unroll 4
        for (int c4 = 0; c4 < C / 4; ++c4) {
            float4 wv = w[c4];
            s += xs[c4 * 4 + 0] * wv.x + xs[c4 * 4 + 1] * wv.y +
                 xs[c4 * 4 + 2] * wv.z + xs[c4 * 4 + 3] * wv.w;
        }
        emb[(size_t)q * C + t] = s;
    }
}

// ---------------------------------------------------------------------------
// Kernel: zero the cluster-memory accumulator (ws is poisoned by harness).
// ---------------------------------------------------------------------------
__global__ void zero_kernel(float* __restrict__ p, int n) {
    int i = blockIdx.x * blockDim.x + threadIdx.x;
    if (i < n) p[i] = 0.0f;
}

// ---------------------------------------------------------------------------
// Kernel 2 (dominant): fused logits GEMM (V_WMMA_F32_16X16X4_F32) + row argmax
// + atomic scatter-sum of point features into cluster memory.
// One block = 32 points x 128 queries. 8 waves; each wave owns one 16-query
// column tile and BOTH 16-row tiles, so each B fragment feeds two independent
// WMMAs (no RAW hazard, halves B-load traffic per WMMA).
// ---------------------------------------------------------------------------
__global__ __launch_bounds__(256) void logits_argmax_scatter_kernel(
    const float* __restrict__ P,        // [N, C] point features
    const float* __restrict__ E,        // [QN, C] mask embeddings
    float* __restrict__ out_logits,     // [N, QN]
    float* __restrict__ cm,             // [QN, C] cluster memory accumulator
    long long N)
{
    __shared__ float a_tile[MB * ASTRIDE];   // staged point tile (padded rows)
    __shared__ float lg[MB * QN];            // logits tile
    __shared__ int assign_s[MB];

    const int t = threadIdx.x;
    const long long m0 = (long long)blockIdx.x * MB;

    // ---- stage A: 32 rows x 256 floats into LDS (row-clamped at tail) ----
    for (int i = t; i < MB * (C / 4); i += 256) {
        int row = i >> 6;            // 64 float4s per row
        int c4  = i & 63;
        long long rg = m0 + row;
        if (rg >= N) rg = N - 1;     // clamp (tail blocks): value unused for stores
        const float* gp = P + rg * C + c4 * 4;
        float* lp = &a_tile[row * ASTRIDE + c4 * 4];
#ifdef USE_ASYNC_LDS
        __builtin_amdgcn_global_load_async_to_lds_b128(
            (gptr_v4i)gp, (lptr_v4i)lp, /*offset=*/0, /*cpol=*/0);
#else
        float4 v = *(const float4*)gp;
        lp[0] = v.x; lp[1] = v.y; lp[2] = v.z; lp[3] = v.w;
#endif
    }
#ifdef USE_ASYNC_LDS
    __builtin_amdgcn_s_wait_asynccnt(0);
#endif
    __syncthreads();

    // ---- WMMA: per-wave two 16x16 logit tiles, K = 256 in steps of 4 ----
    const int lane  = t & 31;
    const int wave  = t >> 5;
    const int q0    = wave * 16;          // this wave's query-column tile
    const int lrow  = lane & 15;          // M (for A) / N (for B) index
    const int khalf = (lane >> 4) * 2;    // lanes 16-31 hold K+2,K+3

    const float* a0row = &a_tile[lrow * ASTRIDE + khalf];
    const float* a1row = &a_tile[(16 + lrow) * ASTRIDE + khalf];
    const float* brow  = E + (size_t)(q0 + lrow) * C + khalf;

    v8f acc0 = {};
    v8f acc1 = {};
    #pragma unroll 4
    for (int k = 0; k < C; k += 4) {
        v2f b  = *(const v2f*)(brow + k);   // B[4x16] frag: lane=N, half=K-pair
        v2f x0 = *(const v2f*)(a0row + k);  // A[16x4] frag (rows 0-15)
        v2f x1 = *(const v2f*)(a1row + k);  // A[16x4] frag (rows 16-31)
        acc0 = __builtin_amdgcn_wmma_f32_16x16x4_f32(
            false, x0, false, b, (short)0, acc0, false, false);
        acc1 = __builtin_amdgcn_wmma_f32_16x16x4_f32(
            false, x1, false, b, (short)0, acc1, false, false);
    }

    // ---- spill accumulators (C/D layout: VGPR r, lane L -> M=r+8*(L/16), N=L%16)
    {
        const int mhi = (lane >> 4) * 8;
        const int col = q0 + lrow;
        #pragma unroll
        for (int r = 0; r < 8; ++r) {
            lg[(r + mhi) * QN + col]        = acc0[r];
            lg[(16 + r + mhi) * QN + col]   = acc1[r];
        }
    }
    __syncthreads();

    // ---- coalesced logits store ----
    for (int i = t; i < MB * QN; i += 256) {
        int row = i >> 7;
        int col = i & (QN - 1);
        long long rg = m0 + row;
        if (rg < N) out_logits[rg * QN + col] = lg[i];
    }

    // ---- argmax per row (first-max tie rule, matching jnp.argmax) ----
    if (t < MB) {
        const float* rowp = &lg[t * QN];
        float best = rowp[0];
        int bi = 0;
        #pragma unroll 4
        for (int q = 1; q < QN; ++q) {
            float v = rowp[q];
            if (v > best) { best = v; bi = q; }
        }
        assign_s[t] = bi;
    }
    __syncthreads();

    // ---- scatter-sum: cm[assign[row], :] += P[row, :] (accumulator is L2-hot)
    #pragma unroll 4
    for (int row = 0; row < MB; ++row) {
        if (m0 + row < N) {
            atomicAdd(&cm[(size_t)assign_s[row] * C + t], a_tile[row * ASTRIDE + t]);
        }
    }
}

// ---------------------------------------------------------------------------
// Kernel 3: bottleneck. LN -> Linear(no bias) -> LN, residual add.
// One block (256 threads) per cluster row.
// ---------------------------------------------------------------------------
__global__ __launch_bounds__(256) void bottleneck_kernel(
    const float* __restrict__ cm,
    const float* __restrict__ centers,
    const float* __restrict__ lnb1w, const float* __restrict__ lnb1b,
    const float* __restrict__ Wb,
    const float* __restrict__ lnb2w, const float* __restrict__ lnb2b,
    float* __restrict__ out2)
{
    __shared__ float xs[C];
    __shared__ float red[C];
    const int t = threadIdx.x;
    const int q = blockIdx.x;

    float x = cm[(size_t)q * C + t];
    float mean = block_sum(x, red) * (1.0f / C);
    float d = x - mean;
    float var = block_sum(d * d, red) * (1.0f / C);
    xs[t] = d * rsqrtf(var + EPS) * lnb1w[t] + lnb1b[t];
    __syncthreads();

    float s = 0.0f;
    const float4* w = (const float4*)(Wb + (size_t)t * C);
    #pragma unroll 4
    for (int c4 = 0; c4 < C / 4; ++c4) {
        float4 wv = w[c4];
        s += xs[c4 * 4 + 0] * wv.x + xs[c4 * 4 + 1] * wv.y +
             xs[c4 * 4 + 2] * wv.z + xs[c4 * 4 + 3] * wv.w;
    }

    float m2 = block_sum(s, red) * (1.0f / C);
    float d2 = s - m2;
    float v2 = block_sum(d2 * d2, red) * (1.0f / C);
    out2[(size_t)q * C + t] =
        centers[(size_t)q * C + t] + d2 * rsqrtf(v2 + EPS) * lnb2w[t] + lnb2b[t];
}

// ---------------------------------------------------------------------------
// Host launcher
// ---------------------------------------------------------------------------
extern "C" void kernel_launch(void* const* d_in, const int* in_sizes, int n_in,
                              void* d_out, int out_size, void* d_ws, size_t ws_size,
                              hipStream_t stream) {
    const float* centers = (const float*)d_in[0];
    const float* P       = (const float*)d_in[1];
    const float* ln0w    = (const float*)d_in[2];
    const float* ln0b    = (const float*)d_in[3];
    const float* W1      = (const float*)d_in[4];
    const float* b1      = (const float*)d_in[5];
    const float* W2      = (const float*)d_in[6];
    const float* b2      = (const float*)d_in[7];
    const float* W3      = (const float*)d_in[8];
    const float* b3      = (const float*)d_in[9];
    const float* lnb1w   = (const float*)d_in[10];
    const float* lnb1b   = (const float*)d_in[11];
    const float* Wb      = (const float*)d_in[12];
    const float* lnb2w   = (const float*)d_in[13];
    const float* lnb2b   = (const float*)d_in[14];

    const long long N = (long long)in_sizes[1] / C;   // 500000

    float* out        = (float*)d_out;
    float* out_logits = out;                          // [N, QN]
    float* out_y      = out + (size_t)N * QN;         // [QN, C]

    float* emb = (float*)d_ws;                        // [QN, C] mask embeddings
    float* cm  = emb + (size_t)QN * C;                // [QN, C] cluster memory

    // 1) MLP head -> mask embeddings
    mlp_head_kernel<<<QN, 256, 0, stream>>>(centers, ln0w, ln0b,
                                            W1, b1, W2, b2, W3, b3, emb);

    // 2) zero cluster-memory accumulator
    zero_kernel<<<(QN * C + 255) / 256, 256, 0, stream>>>(cm, QN * C);

    // 3) fused logits GEMM (WMMA f32) + argmax + scatter-sum
    const int nblocks = (int)((N + MB - 1) / MB);
    logits_argmax_scatter_kernel<<<nblocks, 256, 0, stream>>>(P, emb, out_logits, cm, N);

    // 4) bottleneck + residual
    bottleneck_kernel<<<QN, 256, 0, stream>>>(cm, centers, lnb1w, lnb1b, Wb,
                                              lnb2w, lnb2b, out_y);
}